// DGCNN_16106127360520
// MI455X (gfx1250) — compile-verified
//
#include <hip/hip_runtime.h>
#include <math.h>

// ---------------- problem constants (from reference) ----------------
#define NODES   256                    // nodes per graph
#define BGR     512                    // graphs
#define NTOT    (BGR * NODES)          // 131072
#define DEGK    32                     // edges per node
#define EPG     (NODES * DEGK)         // 8192 edges per graph
#define ETOT    (NTOT * DEGK)          // 4194304 edges
#define XC_LD   132                    // padded row stride of concat buffer (129 -> 132, 16B aligned rows)
#define DTOT    129
#define KPOOL   130
#define FLATSZ  1952
#define DENSE   128

typedef float v2f __attribute__((ext_vector_type(2)));
typedef float v8f __attribute__((ext_vector_type(8)));

// ---------------- degree: deg[i] = 1 + #(src == i), per graph in LDS ----------------
__global__ __launch_bounds__(NODES) void deg_kernel(const int* __restrict__ ei,
                                                    float* __restrict__ degv) {
    __shared__ float cnt[NODES];
    const int g = blockIdx.x, t = threadIdx.x;
    cnt[t] = 1.0f;               // self loop
    __syncthreads();
    const int base = g * EPG, noff = g * NODES;
    for (int e = t; e < EPG; e += NODES)
        atomicAdd(&cnt[ei[base + e] - noff], 1.0f);
    __syncthreads();
    degv[noff + t] = cnt[t];
}

// ---------------- conv layer, Nout=32: hl = A@W+b (WMMA f32), LDS scatter-add, tanh ----------------
// one workgroup per graph, 256 threads = 8 waves; wave w owns rows [32w, 32w+32)
__global__ __launch_bounds__(NODES) void conv32_kernel(
    const float* A, int lda, int acol, int kin,
    const float* __restrict__ W,  const float* __restrict__ bias,
    const int*   __restrict__ ei, const float* __restrict__ degv,
    float* xc, int ocol) {
    extern __shared__ float lds[];
    float* hl  = lds;                 // 256*32 f32
    float* agg = lds + NODES * 32;    // 256*32 f32

    const int g = blockIdx.x, t = threadIdx.x;
    const int lid = t & 31, wv = t >> 5;
    const int m  = lid & 15;          // row within 16-tile (A) / col (B, C)
    const int hi = lid >> 4;          // lane half
    const int kh = hi * 2;            // K sub-offset within 4-step

    v8f acc[2][2] = {};
    const float* Arow0 = A + (size_t)(g * NODES + wv * 32) * lda + acol;
    for (int k0 = 0; k0 < kin; k0 += 4) {
        v2f a0 = *(const v2f*)(Arow0 + (size_t)m        * lda + k0 + kh);
        v2f a1 = *(const v2f*)(Arow0 + (size_t)(16 + m) * lda + k0 + kh);
        v2f b0, b1;
        b0.x = W[(k0 + kh) * 32 + m];          b0.y = W[(k0 + kh + 1) * 32 + m];
        b1.x = W[(k0 + kh) * 32 + 16 + m];     b1.y = W[(k0 + kh + 1) * 32 + 16 + m];
        acc[0][0] = __builtin_amdgcn_wmma_f32_16x16x4_f32(false, a0, false, b0, (short)0, acc[0][0], false, false);
        acc[0][1] = __builtin_amdgcn_wmma_f32_16x16x4_f32(false, a0, false, b1, (short)0, acc[0][1], false, false);
        acc[1][0] = __builtin_amdgcn_wmma_f32_16x16x4_f32(false, a1, false, b0, (short)0, acc[1][0], false, false);
        acc[1][1] = __builtin_amdgcn_wmma_f32_16x16x4_f32(false, a1, false, b1, (short)0, acc[1][1], false, false);
    }
    // C/D layout: VGPR r -> row = r + hi*8, col = lid&15
    #pragma unroll
    for (int rt = 0; rt < 2; ++rt)
        #pragma unroll
        for (int ct = 0; ct < 2; ++ct)
            #pragma unroll
            for (int r = 0; r < 8; ++r) {
                int row = wv * 32 + rt * 16 + hi * 8 + r;
                int col = ct * 16 + m;
                hl[row * 32 + col] = acc[rt][ct][r] + bias[col];
            }
    __syncthreads();

    // self-loop initialization
    #pragma unroll 4
    for (int c = 0; c < 32; ++c) agg[t * 32 + c] = hl[t * 32 + c];
    __syncthreads();

    // edge scatter-add: each wave services one edge per step, 32 channels across lanes
    const int base = g * EPG, noff = g * NODES;
    for (int w = t; w < EPG * 32; w += NODES) {
        int e = w >> 5, c = w & 31;                 // e uniform across the wave
        int s = ei[base + e] - noff;
        int d = ei[ETOT + base + e] - noff;
        atomicAdd(&agg[d * 32 + c], hl[s * 32 + c]);
    }
    __syncthreads();

    const float dinv = 1.0f / degv[noff + t];
    #pragma unroll 4
    for (int c = 0; c < 32; ++c)
        xc[(size_t)(noff + t) * XC_LD + ocol + c] = tanhf(agg[t * 32 + c] * dinv);
}

// ---------------- conv layer 5, Nout=1 (scalar dot), writes xc column 128 ----------------
__global__ __launch_bounds__(NODES) void conv5_kernel(
    const float* __restrict__ W, const float* __restrict__ bias,
    const int* __restrict__ ei, const float* __restrict__ degv, float* xc) {
    __shared__ float hl[NODES];
    __shared__ float agg[NODES];
    const int g = blockIdx.x, t = threadIdx.x, noff = g * NODES;
    const float* row = xc + (size_t)(noff + t) * XC_LD + 96;   // h4 columns
    float s = bias[0];
    #pragma unroll
    for (int k = 0; k < 32; ++k) s += row[k] * W[k];
    hl[t] = s;
    __syncthreads();
    agg[t] = hl[t];
    __syncthreads();
    const int base = g * EPG;
    for (int e = t; e < EPG; e += NODES)
        atomicAdd(&agg[ei[ETOT + base + e] - noff], hl[ei[base + e] - noff]);
    __syncthreads();
    xc[(size_t)(noff + t) * XC_LD + 128] = tanhf(agg[t] / degv[noff + t]);
}

// ---------------- sort-pool (bitonic 256) + c1 + maxpool + c2, per graph ----------------
__global__ __launch_bounds__(NODES) void sortconv_kernel(
    const float* __restrict__ xc,
    const float* __restrict__ wc1, const float* __restrict__ bc1,
    const float* __restrict__ wc2, const float* __restrict__ bc2,
    float* __restrict__ flat) {
    __shared__ float key[NODES];
    __shared__ int   sidx[NODES];
    __shared__ float c1[16 * KPOOL];
    __shared__ float p[16 * 65];
    const int g = blockIdx.x, t = threadIdx.x, noff = g * NODES;
    key[t]  = xc[(size_t)(noff + t) * XC_LD + 128];
    sidx[t] = t;
    __syncthreads();
    // bitonic sort, descending overall
    for (int ksz = 2; ksz <= NODES; ksz <<= 1)
        for (int j = ksz >> 1; j > 0; j >>= 1) {
            int partner = t ^ j;
            if (partner > t) {
                bool ascending = (t & ksz) != 0;
                float a = key[t], b = key[partner];
                bool doSwap = ascending ? (a > b) : (a < b);
                if (doSwap) {
                    key[t] = b; key[partner] = a;
                    int tmp = sidx[t]; sidx[t] = sidx[partner]; sidx[partner] = tmp;
                }
            }
            __syncthreads();
        }
    // c1: [16][130], per-node dense over 129 channels
    for (int u = t; u < 16 * KPOOL; u += NODES) {
        int o = u / KPOOL, kk = u % KPOOL;
        const float* row = xc + (size_t)(noff + sidx[kk]) * XC_LD;
        float s = bc1[o];
        for (int d = 0; d < DTOT; ++d) s += row[d] * wc1[o * DTOT + d];
        c1[o * KPOOL + kk] = fmaxf(s, 0.0f);
    }
    __syncthreads();
    // maxpool(2,2): [16][65]
    for (int u = t; u < 16 * 65; u += NODES) {
        int o = u / 65, j = u % 65;
        p[u] = fmaxf(c1[o * KPOOL + 2 * j], c1[o * KPOOL + 2 * j + 1]);
    }
    __syncthreads();
    // c2: conv1d(16->32, k=5), relu, channel-major flatten
    for (int u = t; u < 32 * 61; u += NODES) {
        int oc = u / 61, j = u % 61;
        float s = bc2[oc];
        #pragma unroll
        for (int ic = 0; ic < 16; ++ic)
            #pragma unroll
            for (int tt = 0; tt < 5; ++tt)
                s += p[ic * 65 + j + tt] * wc2[oc * 80 + ic * 5 + tt];
        flat[(size_t)g * FLATSZ + u] = fmaxf(s, 0.0f);
    }
}

// ---------------- d1 = relu(flat[512x1952] @ Wd1[1952x128] + b), f32 WMMA ----------------
__global__ __launch_bounds__(256) void d1_kernel(
    const float* __restrict__ flat, const float* __restrict__ Wd1,
    const float* __restrict__ bd1, float* __restrict__ d1) {
    const int t = threadIdx.x, lid = t & 31, wv = t >> 5;
    const int tile = blockIdx.x * 8 + wv;      // 256 tiles total
    const int mt = tile >> 3, nt = tile & 7;   // 32 x 8 tiles of 16x16
    const int m = lid & 15, hi = lid >> 4, kh = hi * 2;
    v8f acc = {};
    const float* Arow = flat + (size_t)(mt * 16 + m) * FLATSZ;
    for (int k0 = 0; k0 < FLATSZ; k0 += 4) {
        v2f a = *(const v2f*)(Arow + k0 + kh);
        v2f b;
        b.x = Wd1[(size_t)(k0 + kh)     * DENSE + nt * 16 + m];
        b.y = Wd1[(size_t)(k0 + kh + 1) * DENSE + nt * 16 + m];
        acc = __builtin_amdgcn_wmma_f32_16x16x4_f32(false, a, false, b, (short)0, acc, false, false);
    }
    #pragma unroll
    for (int r = 0; r < 8; ++r) {
        int row = mt * 16 + hi * 8 + r;
        int col = nt * 16 + m;
        d1[row * DENSE + col] = fmaxf(acc[r] + bd1[col], 0.0f);
    }
}

// ---------------- out = d1 @ Wd2[128x6] + b2 ----------------
__global__ void out_kernel(const float* __restrict__ d1, const float* __restrict__ Wd2,
                           const float* __restrict__ bd2, float* __restrict__ out) {
    int i = blockIdx.x * blockDim.x + threadIdx.x;
    if (i >= BGR * 6) return;
    int gph = i / 6, o = i % 6;
    float s = bd2[o];
    #pragma unroll 8
    for (int k = 0; k < DENSE; ++k) s += d1[gph * DENSE + k] * Wd2[k * 6 + o];
    out[i] = s;
}

extern "C" void kernel_launch(void* const* d_in, const int* in_sizes, int n_in,
                              void* d_out, int out_size, void* d_ws, size_t ws_size,
                              hipStream_t stream) {
    const float* x   = (const float*)d_in[0];
    const int*   ei  = (const int*)  d_in[1];
    // d_in[2]=batch, d_in[3]=num_graphs (unused; structure is fixed)
    const float* W1  = (const float*)d_in[4];
    const float* W2  = (const float*)d_in[5];
    const float* W3  = (const float*)d_in[6];
    const float* W4  = (const float*)d_in[7];
    const float* W5  = (const float*)d_in[8];
    const float* b1  = (const float*)d_in[9];
    const float* b2  = (const float*)d_in[10];
    const float* b3  = (const float*)d_in[11];
    const float* b4  = (const float*)d_in[12];
    const float* b5  = (const float*)d_in[13];
    const float* wc1 = (const float*)d_in[14];
    const float* bc1 = (const float*)d_in[15];
    const float* wc2 = (const float*)d_in[16];
    const float* bc2 = (const float*)d_in[17];
    const float* wd1 = (const float*)d_in[18];
    const float* bd1 = (const float*)d_in[19];
    const float* wd2 = (const float*)d_in[20];
    const float* bd2 = (const float*)d_in[21];
    float* out = (float*)d_out;

    // workspace layout (floats): xc | deg | flat | d1   (~74 MB total)
    float* xc   = (float*)d_ws;
    float* degv = xc   + (size_t)NTOT * XC_LD;
    float* flat = degv + NTOT;
    float* d1   = flat + (size_t)BGR * FLATSZ;

    const size_t convLds = (size_t)NODES * 32 * 2 * sizeof(float);  // hl + agg = 64 KB

    deg_kernel   <<<BGR, NODES, 0,       stream>>>(ei, degv);
    conv32_kernel<<<BGR, NODES, convLds, stream>>>(x,  128,   0, 128, W1, b1, ei, degv, xc, 0);
    conv32_kernel<<<BGR, NODES, convLds, stream>>>(xc, XC_LD, 0,  32, W2, b2, ei, degv, xc, 32);
    conv32_kernel<<<BGR, NODES, convLds, stream>>>(xc, XC_LD, 32, 32, W3, b3, ei, degv, xc, 64);
    conv32_kernel<<<BGR, NODES, convLds, stream>>>(xc, XC_LD, 64, 32, W4, b4, ei, degv, xc, 96);
    conv5_kernel <<<BGR, NODES, 0,       stream>>>(W5, b5, ei, degv, xc);
    sortconv_kernel<<<BGR, NODES, 0,     stream>>>(xc, wc1, bc1, wc2, bc2, flat);
    d1_kernel    <<<32, 256, 0,          stream>>>(flat, wd1, bd1, d1);
    out_kernel   <<<(BGR * 6 + 255) / 256, 256, 0, stream>>>(d1, wd2, bd2, out);
}